// TrackSSM_85117661872149
// MI455X (gfx1250) — compile-verified
//
#include <hip/hip_runtime.h>
#include <hip/hip_bf16.h>

typedef __attribute__((ext_vector_type(2))) float v2f;
typedef __attribute__((ext_vector_type(4))) float v4f;
typedef __attribute__((ext_vector_type(8))) float v8f;

#define D_MODEL  256
#define D_STATE  16
#define DT_RANK  16
#define SEQ_L    64
#define N_TRK    256

#define XD_STRIDE 52   // 48 cols padded; 52*4B = 208B keeps rows 16B-aligned, kills bank conflicts
#define Y_STRIDE  260  // 256 cols padded: row*260 -> 4*row mod 64, conflict-free A-frag reads
#define Y_CHUNK   32

__device__ __forceinline__ v2f lo2(v4f v) { return (v2f){v.x, v.y}; }
__device__ __forceinline__ v2f hi2(v4f v) { return (v2f){v.z, v.w}; }

__global__ __launch_bounds__(256)
void track_ssm_fused_kernel(const float* __restrict__ pos,      // (L, B, D)
                            const float* __restrict__ flow,     // (L, B, D)
                            const float* __restrict__ h0,       // (B, D, N)
                            const float* __restrict__ x_proj_w, // (48, D)
                            const float* __restrict__ dt_w,     // (D, 16)
                            const float* __restrict__ dt_b,     // (D,)
                            const float* __restrict__ A_log,    // (D, N)
                            const float* __restrict__ Dvec,     // (D,)
                            const float* __restrict__ out_w,    // (E, D)
                            float* __restrict__ out,            // (B, L*D)
                            float* __restrict__ h_out)          // (B, D, N)
{
    __shared__ float sXd[SEQ_L * XD_STRIDE];     // x_dbl: [t][0:16)=delta_lr [16:32)=B [32:48)=C
    __shared__ float sY [Y_CHUNK * Y_STRIDE];    // y chunk for out-projection

    const int b       = blockIdx.x;
    const int tid     = threadIdx.x;
    const int lane    = tid & 31;
    const int wave    = tid >> 5;
    const int lane16  = lane & 15;
    const int halfSel = lane >> 4;               // 0: K=0..1 / rows 0..7 ; 1: K=2..3 / rows 8..15

    // ============ Phase A: x_dbl = flow_b (64x256) @ x_proj_w^T (256x48), f32 WMMA ============
    for (int tl = wave; tl < 12; tl += 8) {
        const int tm = tl / 3;                   // M tile (t)
        const int jn = tl % 3;                   // N tile (j)
        const float* aBase = flow + ((size_t)(16 * tm + lane16) * N_TRK + b) * D_MODEL + 2 * halfSel;
        const float* bBase = x_proj_w + (size_t)(16 * jn + lane16) * D_MODEL + 2 * halfSel;
        v8f acc = {};
        for (int k = 0; k < D_MODEL; k += 4) {
            v2f a  = *(const v2f*)(aBase + k);
            v2f bb = *(const v2f*)(bBase + k);
            acc = __builtin_amdgcn_wmma_f32_16x16x4_f32(false, a, false, bb,
                                                        (short)0, acc, false, false);
        }
        #pragma unroll
        for (int r = 0; r < 8; ++r) {
            const int row = 16 * tm + r + 8 * halfSel;           // 16x16 f32 D-layout
            sXd[row * XD_STRIDE + 16 * jn + lane16] = acc[r];
        }
    }
    __syncthreads();

    // ============ Phase B/C fused: thread d owns column d; packed-f32 scan ============
    const int d = tid;
    v4f dtw4[4];
    #pragma unroll
    for (int q = 0; q < 4; ++q) dtw4[q] = *(const v4f*)(dt_w + d * DT_RANK + 4 * q);
    const float bias = dt_b[d];
    const float Dd   = Dvec[d];

    v2f A2[8], h2[8];
    #pragma unroll
    for (int q = 0; q < 4; ++q) {
        v4f al = *(const v4f*)(A_log + d * D_STATE + 4 * q);
        A2[2*q+0] = (v2f){-__expf(al.x), -__expf(al.y)};
        A2[2*q+1] = (v2f){-__expf(al.z), -__expf(al.w)};
        v4f hv = *(const v4f*)(h0 + ((size_t)b * D_MODEL + d) * D_STATE + 4 * q);
        h2[2*q+0] = lo2(hv);
        h2[2*q+1] = hi2(hv);
    }

    float* outB = out + (size_t)b * (SEQ_L * D_MODEL);

    for (int chunk = 0; chunk < 2; ++chunk) {
        // ---- scan 32 timesteps, y -> LDS chunk ----
        for (int tt = 0; tt < Y_CHUNK; ++tt) {
            const int t = chunk * Y_CHUNK + tt;
            const v4f* xrow4 = (const v4f*)&sXd[t * XD_STRIDE];
            v4f X[12];                                   // 12x ds_load_b128: delta_lr | B | C
            #pragma unroll
            for (int i = 0; i < 12; ++i) X[i] = xrow4[i];

            // delta = softplus(delta_lr . dtw + bias)   (packed FMAs + horizontal add)
            v4f acc4 = X[0] * dtw4[0];
            acc4 = acc4 + X[1] * dtw4[1];
            acc4 = acc4 + X[2] * dtw4[2];
            acc4 = acc4 + X[3] * dtw4[3];
            const float accs = acc4.x + acc4.y + acc4.z + acc4.w + bias;
            const float dlt  = (accs > 20.0f) ? accs : log1pf(__expf(accs));

            const float p  = pos[((size_t)t * N_TRK + b) * D_MODEL + d];
            const float dp = dlt * p;
            v2f y2 = (v2f){p * Dd, 0.0f};
            #pragma unroll
            for (int i = 0; i < 8; ++i) {
                const v4f Bq = X[4 + (i >> 1)];
                const v4f Cq = X[8 + (i >> 1)];
                const v2f Bi = (i & 1) ? hi2(Bq) : lo2(Bq);
                const v2f Ci = (i & 1) ? hi2(Cq) : lo2(Cq);
                v2f g = dlt * A2[i];                      // v_pk_mul_f32
                v2f e;
                e.x = __expf(g.x);                        // TRANS pipe, co-executes
                e.y = __expf(g.y);
                h2[i] = e * h2[i] + dp * Bi;              // v_pk_fma_f32 x2
                y2 = y2 + h2[i] * Ci;                     // v_pk_fma_f32
            }
            sY[tt * Y_STRIDE + d] = y2.x + y2.y;
        }
        __syncthreads();

        // ---- out-projection of chunk: yc (32x256) @ out_w^T (256x256), f32 WMMA ----
        for (int tl = wave; tl < 32; tl += 8) {
            const int tm = tl >> 4;              // local M tile (0..1)
            const int en = tl & 15;              // N tile (0..15)
            const float* aBase = &sY[(16 * tm + lane16) * Y_STRIDE + 2 * halfSel];
            const float* bBase = out_w + (size_t)(16 * en + lane16) * D_MODEL + 2 * halfSel;
            v8f acc = {};
            for (int k = 0; k < D_MODEL; k += 4) {
                v2f a  = *(const v2f*)(aBase + k);
                v2f bb = *(const v2f*)(bBase + k);
                acc = __builtin_amdgcn_wmma_f32_16x16x4_f32(false, a, false, bb,
                                                            (short)0, acc, false, false);
            }
            const int tbase = chunk * Y_CHUNK + 16 * tm;
            #pragma unroll
            for (int r = 0; r < 8; ++r) {
                const int trow = tbase + r + 8 * halfSel;
                outB[trow * D_MODEL + 16 * en + lane16] = acc[r];
            }
        }
        __syncthreads();
    }

    // ---- h_final (vectorized b128 stores) ----
    #pragma unroll
    for (int q = 0; q < 4; ++q) {
        v4f hv = (v4f){h2[2*q].x, h2[2*q].y, h2[2*q+1].x, h2[2*q+1].y};
        *(v4f*)(h_out + ((size_t)b * D_MODEL + d) * D_STATE + 4 * q) = hv;
    }
}

extern "C" void kernel_launch(void* const* d_in, const int* in_sizes, int n_in,
                              void* d_out, int out_size, void* d_ws, size_t ws_size,
                              hipStream_t stream) {
    const float* pos      = (const float*)d_in[0];
    const float* flow     = (const float*)d_in[1];
    const float* h0       = (const float*)d_in[2];
    const float* x_proj_w = (const float*)d_in[3];
    const float* dt_w     = (const float*)d_in[4];
    const float* dt_b     = (const float*)d_in[5];
    const float* A_log    = (const float*)d_in[6];
    const float* Dvec     = (const float*)d_in[7];
    const float* out_w    = (const float*)d_in[8];

    float* out   = (float*)d_out;                               // (B, L*D) flat
    float* h_out = out + (size_t)N_TRK * SEQ_L * D_MODEL;       // (B, D, N) flat

    track_ssm_fused_kernel<<<N_TRK, 256, 0, stream>>>(
        pos, flow, h0, x_proj_w, dt_w, dt_b, A_log, Dvec, out_w, out, h_out);
}